// APPNPModule_88364657148503
// MI455X (gfx1250) — compile-verified
//
#include <hip/hip_runtime.h>
#include <stdint.h>

#define D      64
#define DQ     16          // D/4 float4 quads per row
#define TILE   256         // edges staged per block
#define TPB    256
#define ALPHA  0.1f
#define K_HOPS 10

// ---------------- gfx1250 async global->LDS staging ----------------
__device__ __forceinline__ void async_load_b32_to_lds(unsigned lds_addr, const void* gaddr) {
  // VDST = VGPR holding LDS byte offset, VADDR = 64-bit global address pair.
  asm volatile("global_load_async_to_lds_b32 %0, %1, off"
               :: "v"(lds_addr), "v"(gaddr) : "memory");
}
__device__ __forceinline__ void wait_asynccnt0() {
  asm volatile("s_wait_asynccnt 0" ::: "memory");
}
__device__ __forceinline__ unsigned lds_offset(const void* p) {
  // addrspace(3) generic pointers carry the wave-relative LDS offset in the low 32 bits
  return (unsigned)(size_t)p;
}

// ---------------- setup kernels ----------------
__global__ void zero_f32(float* __restrict__ p, int n) {
  int i = blockIdx.x * blockDim.x + threadIdx.x;
  if (i < n) p[i] = 0.0f;
}

__global__ void deg_kernel(const int* __restrict__ dst, float* __restrict__ deg, int E) {
  int e = blockIdx.x * blockDim.x + threadIdx.x;
  if (e < E) atomicAdd(&deg[dst[e]], 1.0f);
}

// in-place: deg -> dinv; also self_w = dinv*dinv
__global__ void dinv_kernel(float* __restrict__ deg_dinv, float* __restrict__ self_w, int N) {
  int i = blockIdx.x * blockDim.x + threadIdx.x;
  if (i < N) {
    float d = __frsqrt_rn(deg_dinv[i] + 1.0f);
    deg_dinv[i] = d;
    self_w[i]   = d * d;
  }
}

__global__ void edgew_kernel(const int* __restrict__ src, const int* __restrict__ dst,
                             const float* __restrict__ dinv, float* __restrict__ ew, int E) {
  int e = blockIdx.x * blockDim.x + threadIdx.x;
  if (e < E) ew[e] = dinv[src[e]] * dinv[dst[e]];
}

// acc[i,:] = self_w[i] * h[i,:]   (float4 per thread)
__global__ void init_acc_kernel(const float* __restrict__ h, const float* __restrict__ sw,
                                float* __restrict__ acc, int N) {
  int idx = blockIdx.x * blockDim.x + threadIdx.x;   // one quad
  if (idx < N * DQ) {
    float w  = sw[idx >> 4];
    float4 v = reinterpret_cast<const float4*>(h)[idx];
    float4 r = make_float4(v.x * w, v.y * w, v.z * w, v.w * w);
    reinterpret_cast<float4*>(acc)[idx] = r;
  }
}

// ---------------- per-hop edge scatter (the hot kernel) ----------------
// 16 lanes per edge, float4 per lane: one wave32 = 2 edges' full 64-wide rows.
// Edge metadata for the tile staged in LDS via gfx1250 async copies.
__global__ void __launch_bounds__(TPB)
edge_scatter_kernel(const int*   __restrict__ src,
                    const int*   __restrict__ dst,
                    const float* __restrict__ ew,
                    const float* __restrict__ h,
                    float*       __restrict__ acc, int E) {
  __shared__ int   s_src[TILE];
  __shared__ int   s_dst[TILE];
  __shared__ float s_w[TILE];

  const int t    = threadIdx.x;
  const int base = blockIdx.x * TILE;

  if (base + t < E) {
    async_load_b32_to_lds(lds_offset(&s_src[t]), src + base + t);
    async_load_b32_to_lds(lds_offset(&s_dst[t]), dst + base + t);
    async_load_b32_to_lds(lds_offset(&s_w[t]),   ew  + base + t);
  }
  wait_asynccnt0();
  __syncthreads();

  const int quad = t & 15;   // which float4 of the 64-wide row
  const int eg   = t >> 4;   // edge slot within the group of 16

#pragma unroll 4
  for (int it = 0; it < TILE / 16; ++it) {
    const int el = it * 16 + eg;
    const int e  = base + el;
    if (e < E) {
      const int   s = s_src[el];
      const int   d = s_dst[el];
      const float w = s_w[el];
      const float4 v =
          *reinterpret_cast<const float4*>(h + (size_t)s * D + quad * 4);
      float* o = acc + (size_t)d * D + quad * 4;
      atomicAdd(o + 0, v.x * w);   // global_atomic_add_f32, resolved in 192MB L2
      atomicAdd(o + 1, v.y * w);
      atomicAdd(o + 2, v.z * w);
      atomicAdd(o + 3, v.w * w);
    }
  }
}

// h = 0.9*acc + 0.1*x0 ; acc = self_w * h   (re-seed for next hop, in place)
__global__ void mix_update_kernel(const float* __restrict__ x0, const float* __restrict__ sw,
                                  float* __restrict__ acc, float* __restrict__ h, int N) {
  int idx = blockIdx.x * blockDim.x + threadIdx.x;
  if (idx < N * DQ) {
    float w  = sw[idx >> 4];
    float4 a = reinterpret_cast<const float4*>(acc)[idx];
    float4 x = reinterpret_cast<const float4*>(x0)[idx];
    float4 hn = make_float4(fmaf(1.0f - ALPHA, a.x, ALPHA * x.x),
                            fmaf(1.0f - ALPHA, a.y, ALPHA * x.y),
                            fmaf(1.0f - ALPHA, a.z, ALPHA * x.z),
                            fmaf(1.0f - ALPHA, a.w, ALPHA * x.w));
    reinterpret_cast<float4*>(h)[idx] = hn;
    reinterpret_cast<float4*>(acc)[idx] =
        make_float4(w * hn.x, w * hn.y, w * hn.z, w * hn.w);
  }
}

// out = relu(0.9*acc + 0.1*x0)
__global__ void final_kernel(const float* __restrict__ x0, const float* __restrict__ acc,
                             float* __restrict__ out, int N) {
  int idx = blockIdx.x * blockDim.x + threadIdx.x;
  if (idx < N * DQ) {
    float4 a = reinterpret_cast<const float4*>(acc)[idx];
    float4 x = reinterpret_cast<const float4*>(x0)[idx];
    float4 r = make_float4(fmaxf(fmaf(1.0f - ALPHA, a.x, ALPHA * x.x), 0.0f),
                           fmaxf(fmaf(1.0f - ALPHA, a.y, ALPHA * x.y), 0.0f),
                           fmaxf(fmaf(1.0f - ALPHA, a.z, ALPHA * x.z), 0.0f),
                           fmaxf(fmaf(1.0f - ALPHA, a.w, ALPHA * x.w), 0.0f));
    reinterpret_cast<float4*>(out)[idx] = r;
  }
}

// ---------------- host launch ----------------
extern "C" void kernel_launch(void* const* d_in, const int* in_sizes, int n_in,
                              void* d_out, int out_size, void* d_ws, size_t ws_size,
                              hipStream_t stream) {
  const float* x  = (const float*)d_in[0];
  const int*   ei = (const int*)d_in[1];
  const int N = in_sizes[0] / D;
  const int E = in_sizes[1] / 2;
  const int* src = ei;
  const int* dst = ei + E;

  // workspace carving (all offsets 256B-aligned)
  auto align_up = [](size_t v) { return (v + 63) & ~(size_t)63; };  // in floats -> 256B
  float* w = (float*)d_ws;
  float* deg_dinv = w;                        size_t off = align_up((size_t)N);
  float* self_w   = w + off;                  off += align_up((size_t)N);
  float* edge_w   = w + off;                  off += align_up((size_t)E);
  float* acc      = w + off;                  off += (size_t)N * D;
  float* h        = w + off;                  // N*D floats

  const int tpb = TPB;
  const int gN  = (N + tpb - 1) / tpb;
  const int gE  = (E + tpb - 1) / tpb;
  const int gQ  = (N * DQ + tpb - 1) / tpb;
  const int gT  = (E + TILE - 1) / TILE;

  // one-time graph normalization
  zero_f32<<<gN, tpb, 0, stream>>>(deg_dinv, N);
  deg_kernel<<<gE, tpb, 0, stream>>>(dst, deg_dinv, E);
  dinv_kernel<<<gN, tpb, 0, stream>>>(deg_dinv, self_w, N);
  edgew_kernel<<<gE, tpb, 0, stream>>>(src, dst, deg_dinv, edge_w, E);

  // hop 0 gathers straight from the input features
  init_acc_kernel<<<gQ, tpb, 0, stream>>>(x, self_w, acc, N);

  for (int k = 0; k < K_HOPS; ++k) {
    const float* hcur = (k == 0) ? x : h;
    edge_scatter_kernel<<<gT, tpb, 0, stream>>>(src, dst, edge_w, hcur, acc, E);
    if (k + 1 < K_HOPS) {
      mix_update_kernel<<<gQ, tpb, 0, stream>>>(x, self_w, acc, h, N);
    } else {
      final_kernel<<<gQ, tpb, 0, stream>>>(x, acc, (float*)d_out, N);
    }
  }
}